// Capsule_15152644620559
// MI455X (gfx1250) — compile-verified
//
#include <hip/hip_runtime.h>
#include <math.h>

typedef float v2f __attribute__((ext_vector_type(2)));
typedef float v8f __attribute__((ext_vector_type(8)));
typedef unsigned int u32x4 __attribute__((ext_vector_type(4)));
typedef int i32x4 __attribute__((ext_vector_type(4)));
typedef int i32x8 __attribute__((ext_vector_type(8)));

// problem sizes
#define B_SZ   32
#define N_SZ   4096
#define DIN    128
#define JCAP   32
#define DCAP   16
#define KCOL   512   // JCAP*DCAP

// workspace layout (float offsets)
#define G_OFF     0              // 128*128 Gram (atomic)
#define UBAR_OFF  16384          // 32*128 per-b column sums (atomic)
#define SCAL_OFF  20480          // 16 scalars: 0=gsq 1=gsum 2=gscale (atomic)
#define T_OFF     20496          // 32*32*128 t matrices (atomic)
#define ZERO_LEN  151568         // contiguous atomic region to zero
#define S_OFF     151568         // 32*512 sjh2
#define V_OFF     167952         // 32*128*32 folded V matrices
// total = 299024 floats ~= 1.17 MB of d_ws

__device__ inline v8f wmma_f32(v2f a, v2f b, v8f c) {
  // D = A(16x4) x B(4x16) + C, fp32 — V_WMMA_F32_16X16X4_F32
  return __builtin_amdgcn_wmma_f32_16x16x4_f32(false, a, false, b, (short)0, c,
                                               false, false);
}

// TDM: DMA one 64-row x 128-f32 tile (row stride 128 f32 in memory) into LDS
// at byte offset `lds_byte`, inserting 4 pad DWORDs after every 128 DWORDs so
// the LDS image has the bank-conflict-free 132-float row stride.
// D# packing per CDNA5 ISA (2D tensor, groups 2/3
// zero). This toolchain exposes the 6-arg builtin form.
__device__ inline void tdm_load_tile_64x128(const float* gsrc, unsigned lds_byte) {
  unsigned long long ga = (unsigned long long)(size_t)(const void*)gsrc;
  u32x4 g0;
  g0[0] = 1u;                                     // count=1 user descriptor
  g0[1] = lds_byte;                               // lds_addr
  g0[2] = (unsigned)ga;                           // global_addr[31:0]
  g0[3] = (unsigned)((ga >> 32) & 0x01FFFFFFu)    // global_addr[56:32]
          | (2u << 30);                           // type=2 ("image")
  i32x8 g1;
  g1[0] = (int)((2u << 16)        // data_size = 2 -> 4 bytes
                | (1u << 20)      // pad_enable
                | (6u << 22)      // pad_interval code 6 -> every 128 DWORDs
                | (3u << 25));    // pad_amount  code 3 -> 4 DWORDs
  g1[1] = (int)(128u << 16);      // tensor_dim0 = 128 (bits 79:48, low half)
  g1[2] = (int)(64u << 16);       // tensor_dim1 = 64  (bits 111:80, low half)
  g1[3] = (int)(128u << 16);      // tile_dim0 = 128   (bits 127:112)
  g1[4] = 64;                     // tile_dim1 = 64, tile_dim2 = 0
  g1[5] = 128;                    // tensor_dim0_stride = 128 (low 32)
  g1[6] = 0;                      // stride0 hi / stride1 lo
  g1[7] = 0;                      // stride1 hi
  i32x4 gz4 = {0, 0, 0, 0};       // 2D tensor: groups 2/3 unused
  i32x8 gz8 = {0, 0, 0, 0, 0, 0, 0, 0};
  __builtin_amdgcn_tensor_load_to_lds(g0, g1, gz4, gz4, gz8, 0);
}

__global__ __launch_bounds__(256) void k_zero(float* ws) {
  int i = blockIdx.x * 256 + threadIdx.x;
  if (i < ZERO_LEN) ws[i] = 0.f;
}

// G = U^T U (128x128) + per-b column sums. 128 blocks x 1024 rows each.
// (Keeps VGPR loads: the values are needed in-register for the column sums.)
__global__ __launch_bounds__(256) void k_gram(const float* __restrict__ u,
                                              float* __restrict__ ws) {
  __shared__ float lu[64 * 132];           // 64 rows x 128 f32, stride 132
  float* G    = ws + G_OFF;
  float* ubar = ws + UBAR_OFF;
  const int tid  = threadIdx.x;
  const int w    = tid >> 5;
  const int lane = tid & 31;
  const int l16  = lane & 15;
  const int khi  = (lane < 16) ? 0 : 2;    // K offset for A/B VGPR0
  const int mro  = (lane < 16) ? 0 : 8;    // C/D row offset for lane half
  const int rowbase = blockIdx.x * 1024;   // rows within flattened [B*N]
  const int b = rowbase >> 12;             // /4096: block stays within one b
  v8f z = {0.f, 0.f, 0.f, 0.f, 0.f, 0.f, 0.f, 0.f};
  v8f acc[8];
#pragma unroll
  for (int i = 0; i < 8; ++i) acc[i] = z;
  float csum = 0.f;                        // column sum, f = tid&127 fixed

  for (int ch = 0; ch < 16; ++ch) {
    const int base = (rowbase + ch * 64) * DIN;
#pragma unroll 4
    for (int e = 0; e < 32; ++e) {
      int lin = e * 256 + tid;             // f = lin&127 == tid&127 always
      float x = u[base + lin];
      lu[(lin >> 7) * 132 + (lin & 127)] = x;
      csum += x;
    }
    __syncthreads();
    for (int ks = 0; ks < 16; ++ks) {      // 4-row K slices over n
      const int r0 = (ks * 4 + khi) * 132;
      const int r1 = r0 + 132;
      v2f bv; bv.x = lu[r0 + w * 16 + l16]; bv.y = lu[r1 + w * 16 + l16];
#pragma unroll
      for (int i = 0; i < 8; ++i) {
        v2f av; av.x = lu[r0 + i * 16 + l16]; av.y = lu[r1 + i * 16 + l16];
        acc[i] = wmma_f32(av, bv, acc[i]); // G[i-block, w-block] tile
      }
    }
    __syncthreads();
  }
#pragma unroll
  for (int i = 0; i < 8; ++i)
#pragma unroll
    for (int r = 0; r < 8; ++r)
      atomicAdd(&G[(i * 16 + r + mro) * 128 + w * 16 + l16], acc[i][r]);
  atomicAdd(&ubar[b * 128 + (tid & 127)], csum);
}

// gsq = sum_k w_k^T G w_k ; gsum = sum_k (sum_b ubar_b) . w_k
__global__ __launch_bounds__(256) void k_gsq(const float* __restrict__ W,
                                             float* __restrict__ ws) {
  __shared__ float us[128];
  const float* G    = ws + G_OFF;
  const float* ubar = ws + UBAR_OFF;
  float* scal = ws + SCAL_OFF;
  const int k = blockIdx.x * 256 + threadIdx.x;  // 0..511
  if (threadIdx.x < 128) {
    float s = 0.f;
    for (int bb = 0; bb < B_SZ; ++bb) s += ubar[bb * 128 + threadIdx.x];
    us[threadIdx.x] = s;
  }
  __syncthreads();
  float qk = 0.f, sk = 0.f;
  for (int i = 0; i < 128; ++i) {
    float wik = W[i * KCOL + k];
    float gi = 0.f;
    for (int j = 0; j < 128; ++j) gi += G[i * 128 + j] * W[j * KCOL + k];
    qk += wik * gi;
    sk += us[i] * wik;
  }
  atomicAdd(&scal[0], qk);
  atomicAdd(&scal[1], sk);
}

// gscale, sjh2, global normalize, V_b = gscale*inv2*(W @ sjh2)
__global__ __launch_bounds__(256) void k_prep(const float* __restrict__ W,
                                              float* __restrict__ ws) {
  __shared__ float red[256];
  const float* ubar = ws + UBAR_OFF;
  float* scal = ws + SCAL_OFF;
  float* S = ws + S_OFF;
  float* V = ws + V_OFF;
  const int tid = threadIdx.x;
  const float gscale = scal[1] * rsqrtf(fmaxf(scal[0], 1e-12f));
  float part = 0.f;
  for (int out = tid; out < B_SZ * KCOL; out += 256) {
    int bb = out >> 9, k = out & 511;
    float s = 0.f;
    for (int i = 0; i < 128; ++i) s += ubar[bb * 128 + i] * W[i * KCOL + k];
    float sj = gscale * 0.03125f * s;      // (gscale/32) * S
    S[out] = sj;
    part += sj * sj;
  }
  red[tid] = part;
  __syncthreads();
  for (int st = 128; st > 0; st >>= 1) {
    if (tid < st) red[tid] += red[tid + st];
    __syncthreads();
  }
  const float c = gscale * rsqrtf(fmaxf(red[0], 1e-12f));
  for (int out = tid; out < B_SZ * DIN * JCAP; out += 256) {
    int bb = out >> 12, rem = out & 4095, i = rem >> 5, j = rem & 31;
    float a = 0.f;
#pragma unroll
    for (int d = 0; d < DCAP; ++d)
      a += W[i * KCOL + j * 16 + d] * S[bb * KCOL + j * 16 + d];
    V[out] = c * a;                        // logits scale folded in
  }
  if (tid == 0) scal[2] = gscale;
}

// fused: logits = U_b V_b (WMMA), softmax over j, t_b += C^T U_b (WMMA)
// U tiles staged LDS-direct by the Tensor Data Mover (TENSORcnt tracked).
__global__ __launch_bounds__(256) void k_route(const float* __restrict__ u,
                                               float* __restrict__ ws) {
  __shared__ float lu[64 * 132];  // U chunk: 64 n x 128 f (TDM pads to 132)
  __shared__ float lv[128 * 33];  // V_b:     128 f x 32 j
  __shared__ float lc[64 * 33];   // logits -> cij: 64 n x 32 j
  const float* V = ws + V_OFF;
  float* T = ws + T_OFF;
  const int tid  = threadIdx.x;
  const int w    = tid >> 5;
  const int lane = tid & 31;
  const int l16  = lane & 15;
  const int khi  = (lane < 16) ? 0 : 2;
  const int mro  = (lane < 16) ? 0 : 8;
  const int b     = blockIdx.x >> 3;
  const int nbase = (blockIdx.x & 7) * 512;
  const unsigned lu_byte = (unsigned)(size_t)(void*)lu;  // LDS byte offset
  for (int lin = tid; lin < DIN * JCAP; lin += 256)
    lv[(lin >> 5) * 33 + (lin & 31)] = V[b * DIN * JCAP + lin];
  v8f z = {0.f, 0.f, 0.f, 0.f, 0.f, 0.f, 0.f, 0.f};
  v8f tacc0 = z, tacc1 = z;
  const int ln0 = (w & 3) * 16;            // logits n-subtile
  const int ljt = (w >> 2) * 16;           // logits / t j-tile
  const int tf0 = ((w & 3) * 2) * 16;      // t feature tiles
  const int tf1 = tf0 + 16;
  __syncthreads();

  for (int ch = 0; ch < 8; ++ch) {
    // ---- TDM stage: wave 0 DMAs the 64x128 tile into LDS with padding ----
    if (w == 0) {
      const int base = (b * N_SZ + nbase + ch * 64) * DIN;
      tdm_load_tile_64x128(u + base, lu_byte);
      __builtin_amdgcn_s_wait_tensorcnt(0);
    }
    __syncthreads();                       // publish LDS tile to all waves
    // ---- logits tile: A = U rows (16n x 4f), B = V (4f x 16j) ----
    v8f lacc = z;
    for (int ks = 0; ks < 32; ++ks) {
      int f0 = ks * 4;
      v2f av, bv;
      av.x = lu[(ln0 + l16) * 132 + f0 + khi];
      av.y = lu[(ln0 + l16) * 132 + f0 + khi + 1];
      bv.x = lv[(f0 + khi) * 33 + ljt + l16];
      bv.y = lv[(f0 + khi + 1) * 33 + ljt + l16];
      lacc = wmma_f32(av, bv, lacc);
    }
#pragma unroll
    for (int r = 0; r < 8; ++r)
      lc[(ln0 + r + mro) * 33 + ljt + l16] = lacc[r];
    __syncthreads();
    // ---- softmax over j (32 values per n-row), rows 0..63 ----
    if (tid < 64) {
      float mx = -3.4e38f;
      for (int j = 0; j < 32; ++j) mx = fmaxf(mx, lc[tid * 33 + j]);
      float s = 0.f;
      for (int j = 0; j < 32; ++j) {
        float e = __expf(lc[tid * 33 + j] - mx);
        lc[tid * 33 + j] = e;
        s += e;
      }
      float inv = 1.f / s;
      for (int j = 0; j < 32; ++j) lc[tid * 33 + j] *= inv;
    }
    __syncthreads();
    // ---- t tile: A = C^T (16j x 4n, transposed LDS read), B = U (4n x 16f)
    for (int ks = 0; ks < 16; ++ks) {
      int r0 = ks * 4 + khi;
      v2f av, b0, b1;
      av.x = lc[r0 * 33 + ljt + l16];       av.y = lc[(r0 + 1) * 33 + ljt + l16];
      b0.x = lu[r0 * 132 + tf0 + l16];      b0.y = lu[(r0 + 1) * 132 + tf0 + l16];
      b1.x = lu[r0 * 132 + tf1 + l16];      b1.y = lu[(r0 + 1) * 132 + tf1 + l16];
      tacc0 = wmma_f32(av, b0, tacc0);
      tacc1 = wmma_f32(av, b1, tacc1);
    }
    __syncthreads();                       // all reads done before next DMA
  }
#pragma unroll
  for (int r = 0; r < 8; ++r) {
    int j = ljt + r + mro;
    atomicAdd(&T[b * 4096 + j * 128 + tf0 + l16], tacc0[r]);
    atomicAdd(&T[b * 4096 + j * 128 + tf1 + l16], tacc1[r]);
  }
}

// sjh3 = gscale * (t_b W), squash per (b,j) over d=16. One half-wave per pair.
__global__ __launch_bounds__(256) void k_final(const float* __restrict__ W,
                                               const float* __restrict__ ws,
                                               float* __restrict__ out) {
  const float* T    = ws + T_OFF;
  const float* scal = ws + SCAL_OFF;
  const int lane = threadIdx.x & 31;
  const int wv   = (blockIdx.x * 256 + threadIdx.x) >> 5;  // 0..511
  const int pair = wv * 2 + (lane >> 4);                   // 0..1023 (b,j)
  const int b = pair >> 5, j = pair & 31, d = lane & 15;
  const float gscale = scal[2];
  float a = 0.f;
  for (int i = 0; i < 128; ++i)
    a += T[b * 4096 + j * 128 + i] * W[i * KCOL + j * 16 + d];
  float sjh = gscale * a;
  float s2 = sjh * sjh;                    // reduce over 16 lanes of the half
  s2 += __shfl_xor(s2, 1, 32);
  s2 += __shfl_xor(s2, 2, 32);
  s2 += __shfl_xor(s2, 4, 32);
  s2 += __shfl_xor(s2, 8, 32);
  s2 += 1e-7f;
  out[b * KCOL + j * 16 + d] = (sqrtf(s2) / (1.f + s2)) * sjh;
}

extern "C" void kernel_launch(void* const* d_in, const int* in_sizes, int n_in,
                              void* d_out, int out_size, void* d_ws,
                              size_t ws_size, hipStream_t stream) {
  (void)in_sizes; (void)n_in; (void)out_size; (void)ws_size;
  const float* u = (const float*)d_in[0];   // [32,4096,128] f32
  const float* W = (const float*)d_in[1];   // [1,128,512]   f32
  float* out = (float*)d_out;               // [32,32,16]    f32
  float* ws  = (float*)d_ws;

  k_zero <<<(ZERO_LEN + 255) / 256, 256, 0, stream>>>(ws);
  k_gram <<<128, 256, 0, stream>>>(u, ws);
  k_gsq  <<<2,   256, 0, stream>>>(W, ws);
  k_prep <<<1,   256, 0, stream>>>(W, ws);
  k_route<<<256, 256, 0, stream>>>(u, ws);
  k_final<<<64,  256, 0, stream>>>(W, ws, out);
}